// Model_74019466379907
// MI455X (gfx1250) — compile-verified
//
#include <hip/hip_runtime.h>
#include <hip/hip_bf16.h>
#include <math.h>

#define Bn 512
#define Sn 200
#define Sp 208              // S padded to a multiple of 16 (13 tiles)
#define Hn 128
#define NEGF (-9.0e15f)
#define ALPHA_LRELU 0.2f

typedef float v2f __attribute__((ext_vector_type(2)));
typedef float v8f __attribute__((ext_vector_type(8)));

__device__ __forceinline__ float lrelu(float x) {
    return x >= 0.0f ? x : ALPHA_LRELU * x;
}

// ---------------------------------------------------------------------------
// Branch-free batched GEMM on the f32 WMMA pipe: C[b] = alpha * A[b] @ op(B[b])
// All dims are multiples of 16 (K multiple of 4); operands are zero-padded by
// their producers, so there are NO bounds checks: EXEC stays all-ones and the
// inner loop is pure loads + v_wmma_f32_16x16x4_f32.
// One wave computes a 16 x (16*NT) strip: the A fragment is loaded once and
// feeds NT back-to-back WMMAs with independent accumulators.
// ---------------------------------------------------------------------------
template <bool TRANSB, int NT>
__global__ __launch_bounds__(32)
void wmma_gemm(const float* __restrict__ A, const float* __restrict__ Bm,
               float* __restrict__ C,
               int K, int lda, int ldb, int ldc,
               long long sA, long long sB, long long sC,
               float alpha) {
    const int b     = blockIdx.z;
    const float* Ab = A  + (long long)b * sA;
    const float* Bb = Bm + (long long)b * sB;
    float*       Cb = C  + (long long)b * sC;

    const int lane = threadIdx.x & 31;
    const int half = lane >> 4;   // 0: K pair {0,1}; 1: K pair {2,3}
    const int l16  = lane & 15;
    const int m0   = blockIdx.y * 16;
    const int n0   = blockIdx.x * (16 * NT);
    const int mA   = m0 + l16;    // A-fragment row (lane-striped M)

    v8f acc[NT];
    #pragma unroll
    for (int t = 0; t < NT; ++t)
        acc[t] = (v8f){0.f, 0.f, 0.f, 0.f, 0.f, 0.f, 0.f, 0.f};

    const float* aRow = Ab + (long long)mA * lda;

    for (int k0 = 0; k0 < K; k0 += 4) {
        const int kb = k0 + half * 2;
        const v2f a = *(const v2f*)(aRow + kb);
        #pragma unroll
        for (int t = 0; t < NT; ++t) {
            const int nB = n0 + t * 16 + l16;
            v2f bv;
            if (TRANSB) {
                bv = *(const v2f*)(Bb + (long long)nB * ldb + kb);
            } else {
                bv.x = Bb[(long long)kb * ldb + nB];
                bv.y = Bb[(long long)(kb + 1) * ldb + nB];
            }
            acc[t] = __builtin_amdgcn_wmma_f32_16x16x4_f32(
                false, a, false, bv, (short)0, acc[t], false, false);
        }
    }

    #pragma unroll
    for (int t = 0; t < NT; ++t) {
        const int nB = n0 + t * 16 + l16;
        #pragma unroll
        for (int r = 0; r < 8; ++r) {
            const int mo = m0 + r + 8 * half;   // D layout: M = r + 8*(lane>=16)
            Cb[(long long)mo * ldc + nB] = alpha * acc[t][r];
        }
    }
}

// ---------------------------------------------------------------------------
// hidden[b,s,:] = emb[items[b,s]] for s<Sn ; 0 for pad rows (Sn..Sp)
// ---------------------------------------------------------------------------
__global__ void k_gather_emb(const int* __restrict__ items,
                             const float* __restrict__ emb,
                             float* __restrict__ hidden) {
    long long i = (long long)blockIdx.x * blockDim.x + threadIdx.x;
    if (i >= (long long)Bn * Sp * Hn) return;
    int h = (int)(i % Hn);
    long long bsp = i / Hn;
    int s = (int)(bsp % Sp);
    long long b = bsp / Sp;
    float v = 0.f;
    if (s < Sn) {
        int it = items[b * Sn + s];
        v = emb[(long long)it * Hn + h];
    }
    hidden[i] = v;
}

// ---------------------------------------------------------------------------
// s[b,i] = <hidden[b,i,:], wa>, t[b,i] = <hidden[b,i,:], wb>  (valid rows only)
// ---------------------------------------------------------------------------
__global__ __launch_bounds__(Hn)
void k_st(const float* __restrict__ hidden, const float* __restrict__ tr_w,
          float* __restrict__ sv, float* __restrict__ tv) {
    int row = blockIdx.x;            // 0 .. B*Sn-1
    int h   = threadIdx.x;
    int b   = row / Sn, s = row % Sn;
    __shared__ float r0[Hn];
    __shared__ float r1[Hn];
    float x = hidden[((long long)b * Sp + s) * Hn + h];
    r0[h] = x * tr_w[h];
    r1[h] = x * tr_w[Hn + h];
    __syncthreads();
    for (int off = Hn / 2; off > 0; off >>= 1) {
        if (h < off) { r0[h] += r0[h + off]; r1[h] += r1[h + off]; }
        __syncthreads();
    }
    if (h == 0) { sv[row] = r0[0]; tv[row] = r1[0]; }
}

// ---------------------------------------------------------------------------
// Symmetrized masked tanh attention row + softmax, into padded (Sp x Sp)
// matrix. Pad rows (i>=Sn) and pad cols (j>=Sn) are written as zero, so
// padded K/M regions contribute nothing to downstream WMMA GEMMs.
// ---------------------------------------------------------------------------
__global__ __launch_bounds__(256)
void k_att(const int* __restrict__ adj, const float* __restrict__ sv,
           const float* __restrict__ tv, const float* __restrict__ trb_p,
           float* __restrict__ A) {
    int i = blockIdx.x, b = blockIdx.y, j = threadIdx.x;
    long long rowo = ((long long)b * Sp + i) * Sp;
    if (i >= Sn) {                       // pad row: all zeros
        if (j < Sp) A[rowo + j] = 0.f;
        return;
    }
    __shared__ float vals[Sn];
    __shared__ float red[256];
    const float trb = trb_p[0];
    const float si = sv[b * Sn + i];
    const float ti = tv[b * Sn + i];
    if (j < Sn) {
        int aij = adj[((long long)b * Sn + i) * Sn + j];
        int aji = adj[((long long)b * Sn + j) * Sn + i];
        float sj = sv[b * Sn + j];
        float tj = tv[b * Sn + j];
        float o1 = (aij == 1) ? tanhf(si + tj + trb) : 0.f;
        float o2 = (aji == 1) ? tanhf(sj + ti + trb) : 0.f;
        float a  = 0.5f * (o1 + o2);
        if (aij + aji == 0) a = NEGF;
        vals[j] = a;
    }
    red[j] = (j < Sn) ? vals[j] : -3.0e38f;
    __syncthreads();
    for (int off = 128; off > 0; off >>= 1) {
        if (j < off) red[j] = fmaxf(red[j], red[j + off]);
        __syncthreads();
    }
    float m = red[0];
    __syncthreads();
    float e = (j < Sn) ? expf(vals[j] - m) : 0.f;
    red[j] = e;
    __syncthreads();
    for (int off = 128; off > 0; off >>= 1) {
        if (j < off) red[j] += red[j + off];
        __syncthreads();
    }
    float inv = 1.f / red[0];
    if (j < Sn) A[rowo + j] = e * inv;
    else if (j < Sp) A[rowo + j] = 0.f;  // pad cols -> zero K contribution
}

// ---------------------------------------------------------------------------
// hidden_g[b,i,:] = hidden[b, alias[b,i], :] ; pad rows zero
// ---------------------------------------------------------------------------
__global__ void k_alias(const int* __restrict__ alias,
                        const float* __restrict__ hidden,
                        float* __restrict__ hidden_g) {
    long long i = (long long)blockIdx.x * blockDim.x + threadIdx.x;
    if (i >= (long long)Bn * Sp * Hn) return;
    int h = (int)(i % Hn);
    long long bsp = i / Hn;
    int s = (int)(bsp % Sp);
    long long b = bsp / Sp;
    float v = 0.f;
    if (s < Sn) {
        int al = alias[b * Sn + s];
        v = hidden[((long long)b * Sp + al) * Hn + h];
    }
    hidden_g[i] = v;
}

// ---------------------------------------------------------------------------
// Depthwise K=3 conv over sequence of pe = pos_table[pos], + bias, lrelu,
// for q/k/v simultaneously; pad rows zero.
// ---------------------------------------------------------------------------
__global__ void k_posconv(const int* __restrict__ pos,
                          const float* __restrict__ ptab,
                          const float* __restrict__ qw, const float* __restrict__ qb,
                          const float* __restrict__ kw, const float* __restrict__ kb,
                          const float* __restrict__ vw, const float* __restrict__ vb,
                          float* __restrict__ pq, float* __restrict__ pk,
                          float* __restrict__ pv) {
    long long i = (long long)blockIdx.x * blockDim.x + threadIdx.x;
    if (i >= (long long)Bn * Sp * Hn) return;
    int h = (int)(i % Hn);
    long long bsp = i / Hn;
    int s = (int)(bsp % Sp);
    long long b = bsp / Sp;
    if (s >= Sn) {
        pq[i] = 0.f; pk[i] = 0.f; pv[i] = 0.f;
        return;
    }
    const int* prow = pos + b * Sn;
    float x0 = (s - 1 >= 0) ? ptab[(long long)prow[s - 1] * Hn + h] : 0.f;
    float x1 = ptab[(long long)prow[s] * Hn + h];
    float x2 = (s + 1 < Sn) ? ptab[(long long)prow[s + 1] * Hn + h] : 0.f;
    pq[i] = lrelu(x0 * qw[h * 3] + x1 * qw[h * 3 + 1] + x2 * qw[h * 3 + 2] + qb[h]);
    pk[i] = lrelu(x0 * kw[h * 3] + x1 * kw[h * 3 + 1] + x2 * kw[h * 3 + 2] + kb[h]);
    pv[i] = lrelu(x0 * vw[h * 3] + x1 * vw[h * 3 + 1] + x2 * vw[h * 3 + 2] + vb[h]);
}

// ---------------------------------------------------------------------------
// In-place masked softmax over valid cols of padded pa (mask==0 -> NEG).
// Pad cols were stored as zero by the GEMM and are left untouched.
// ---------------------------------------------------------------------------
__global__ __launch_bounds__(256)
void k_pa_softmax(const int* __restrict__ gmask, float* __restrict__ pa) {
    int q = blockIdx.x, b = blockIdx.y, k = threadIdx.x;
    __shared__ float vals[Sn];
    __shared__ float red[256];
    long long rowo = ((long long)b * Sp + q) * Sp;
    if (k < Sn) {
        int gm = gmask[((long long)b * Sn + q) * Sn + k];
        vals[k] = (gm == 0) ? NEGF : pa[rowo + k];
    }
    red[k] = (k < Sn) ? vals[k] : -3.0e38f;
    __syncthreads();
    for (int off = 128; off > 0; off >>= 1) {
        if (k < off) red[k] = fmaxf(red[k], red[k + off]);
        __syncthreads();
    }
    float m = red[0];
    __syncthreads();
    float e = (k < Sn) ? expf(vals[k] - m) : 0.f;
    red[k] = e;
    __syncthreads();
    for (int off = 128; off > 0; off >>= 1) {
        if (k < off) red[k] += red[k + off];
        __syncthreads();
    }
    float inv = 1.f / red[0];
    if (k < Sn) pa[rowo + k] = e * inv;
}

// nh = hidden_g + pos_attn (full padded volume; pads are 0+0)
__global__ void k_add(const float* __restrict__ a, const float* __restrict__ b,
                      float* __restrict__ c) {
    long long i = (long long)blockIdx.x * blockDim.x + threadIdx.x;
    if (i >= (long long)Bn * Sp * Hn) return;
    c[i] = a[i] + b[i];
}

// hs = masked mean over S; h_n = nh[b, sum(mask)-1]
__global__ __launch_bounds__(Hn)
void k_hsn(const int* __restrict__ mask, const float* __restrict__ nh,
           float* __restrict__ hs, float* __restrict__ hn) {
    int b = blockIdx.x, h = threadIdx.x;
    float acc = 0.f;
    int sm = 0;
    for (int s = 0; s < Sn; ++s) {
        int mk = mask[b * Sn + s];
        sm += mk;
        acc += (float)mk * nh[((long long)b * Sp + s) * Hn + h];
    }
    hs[b * Hn + h] = acc / (float)sm;
    int last = sm - 1;
    hn[b * Hn + h] = nh[((long long)b * Sp + last) * Hn + h];
}

// mix = hs@W2^T + hn@W3^T + max(hs,hn)@W4^T  (small: plain VALU)
__global__ void k_mix(const float* __restrict__ hs, const float* __restrict__ hn,
                      const float* __restrict__ W2, const float* __restrict__ W3,
                      const float* __restrict__ W4, float* __restrict__ mix) {
    int idx = blockIdx.x * blockDim.x + threadIdx.x;
    if (idx >= Bn * Hn) return;
    int i = idx % Hn, b = idx / Hn;
    float acc = 0.f;
    for (int j = 0; j < Hn; ++j) {
        float a = hs[b * Hn + j];
        float c = hn[b * Hn + j];
        acc += a * W2[i * Hn + j] + c * W3[i * Hn + j] + fmaxf(a, c) * W4[i * Hn + j];
    }
    mix[idx] = acc;
}

// c[b,s] = sum_h lrelu(T + b1 + mix) * w_2
__global__ __launch_bounds__(Hn)
void k_cvec(const float* __restrict__ T, const float* __restrict__ b1,
            const float* __restrict__ mix, const float* __restrict__ w2,
            float* __restrict__ cvec) {
    int bs = blockIdx.x;          // 0 .. B*Sn-1
    int h  = threadIdx.x;
    int b  = bs / Sn, s = bs % Sn;
    __shared__ float red[Hn];
    float v = lrelu(T[((long long)b * Sp + s) * Hn + h] + b1[h] + mix[b * Hn + h]);
    red[h] = v * w2[h];
    __syncthreads();
    for (int off = Hn / 2; off > 0; off >>= 1) {
        if (h < off) red[h] += red[h + off];
        __syncthreads();
    }
    if (h == 0) cvec[bs] = red[0];
}

// out[b,h] = sum_s mask * (c[b,s] + b_2[h]) * hidden_g[b,s,h]
__global__ __launch_bounds__(Hn)
void k_out(const int* __restrict__ mask, const float* __restrict__ cvec,
           const float* __restrict__ b2, const float* __restrict__ hg,
           float* __restrict__ out) {
    int b = blockIdx.x, h = threadIdx.x;
    float bb = b2[h];
    float acc = 0.f;
    for (int s = 0; s < Sn; ++s) {
        int mk = mask[b * Sn + s];
        if (mk) acc += (cvec[b * Sn + s] + bb) * hg[((long long)b * Sp + s) * Hn + h];
    }
    out[b * Hn + h] = acc;
}

// ---------------------------------------------------------------------------
extern "C" void kernel_launch(void* const* d_in, const int* in_sizes, int n_in,
                              void* d_out, int out_size, void* d_ws, size_t ws_size,
                              hipStream_t stream) {
    (void)in_sizes; (void)n_in; (void)out_size; (void)ws_size;
    const int*   mask   = (const int*)d_in[0];
    const int*   adj    = (const int*)d_in[1];
    const int*   items  = (const int*)d_in[2];
    const int*   alias  = (const int*)d_in[3];
    const int*   pos    = (const int*)d_in[4];
    const int*   gmask  = (const int*)d_in[5];
    const float* emb    = (const float*)d_in[6];
    const float* ptab   = (const float*)d_in[7];
    const float* tr_w   = (const float*)d_in[8];
    const float* tr_b   = (const float*)d_in[9];
    const float* W1     = (const float*)d_in[10];
    const float* b1     = (const float*)d_in[11];
    const float* W2     = (const float*)d_in[12];
    const float* W3     = (const float*)d_in[13];
    const float* W4     = (const float*)d_in[14];
    const float* w_2    = (const float*)d_in[15];
    const float* b_2    = (const float*)d_in[16];
    const float* cq_w   = (const float*)d_in[17];
    const float* cq_b   = (const float*)d_in[18];
    const float* ck_w   = (const float*)d_in[19];
    const float* ck_b   = (const float*)d_in[20];
    const float* cv_w   = (const float*)d_in[21];
    const float* cv_b   = (const float*)d_in[22];
    float* out = (float*)d_out;

    const size_t BSHp = (size_t)Bn * Sp * Hn;     // padded (B,Sp,H)
    const size_t BSSp = (size_t)Bn * Sp * Sp;     // padded (B,Sp,Sp)
    const size_t BS   = (size_t)Bn * Sn;
    const size_t BH   = (size_t)Bn * Hn;

    float* ws      = (float*)d_ws;
    float* hidden0 = ws;
    float* hidden1 = hidden0 + BSHp;
    float* Abuf    = hidden1 + BSHp;        // (B,Sp,Sp): graph attn, then pa
    float* sv      = Abuf + BSSp;
    float* tv      = sv + BS;
    float* pq      = tv + BS;               // p_q, then pos_attn
    float* pk      = pq + BSHp;             // p_k, then nh
    float* pv      = pk + BSHp;             // p_v, then T = nh@W1^T
    float* hs      = pv + BSHp;
    float* hn      = hs + BH;
    float* mixb    = hn + BH;
    float* cvec    = mixb + BH;

    const int ew_blocks = (int)((BSHp + 255) / 256);
    const dim3 attGrid(Sp, Bn);                      // incl. pad rows (zeroed)
    const dim3 smGrid(Sn, Bn);                       // valid rows only
    // M=Sp(13 tiles), N=Hn(128): NT=4 -> 2 strips of 64 cols
    const dim3 gemmMHN(Hn / 64, Sp / 16, Bn);
    // M=Sp, N=Sp(13 tiles): NT=1
    const dim3 gemmMSN(Sp / 16, Sp / 16, Bn);

    // 1. hidden = emb[items] (pad rows zeroed)
    k_gather_emb<<<ew_blocks, 256, 0, stream>>>(items, emb, hidden0);

    // 2. two graph-attention layers: hidden0 -> hidden1 -> hidden0
    float* hin = hidden0;
    float* hout = hidden1;
    for (int layer = 0; layer < 2; ++layer) {
        k_st<<<(int)BS, Hn, 0, stream>>>(hin, tr_w, sv, tv);
        k_att<<<attGrid, 256, 0, stream>>>(adj, sv, tv, tr_b, Abuf);
        wmma_gemm<false, 4><<<gemmMHN, 32, 0, stream>>>(
            Abuf, hin, hout, Sp, Sp, Hn, Hn,
            (long long)Sp * Sp, (long long)Sp * Hn, (long long)Sp * Hn, 1.0f);
        float* t = hin; hin = hout; hout = t;
    }
    // final hidden in hidden0 (hin)

    // 3. alias gather -> hidden_g (into hidden1; pad rows zeroed)
    k_alias<<<ew_blocks, 256, 0, stream>>>(alias, hin, hidden1);
    float* hidden_g = hidden1;

    // 4. positional depthwise conv q/k/v (pad rows zeroed)
    k_posconv<<<ew_blocks, 256, 0, stream>>>(pos, ptab, cq_w, cq_b, ck_w, ck_b,
                                             cv_w, cv_b, pq, pk, pv);

    // 5. pa = p_q @ p_k^T / sqrt(S)   (K=Hn=128)
    wmma_gemm<true, 1><<<gemmMSN, 32, 0, stream>>>(
        pq, pk, Abuf, Hn, Hn, Hn, Sp,
        (long long)Sp * Hn, (long long)Sp * Hn, (long long)Sp * Sp,
        1.0f / sqrtf((float)Sn));

    // 6. masked softmax (valid 200x200 region)
    k_pa_softmax<<<smGrid, 256, 0, stream>>>(gmask, Abuf);

    // 7. pos_attn = pa @ p_v  (into pq; K=Sp)
    wmma_gemm<false, 4><<<gemmMHN, 32, 0, stream>>>(
        Abuf, pv, pq, Sp, Sp, Hn, Hn,
        (long long)Sp * Sp, (long long)Sp * Hn, (long long)Sp * Hn, 1.0f);

    // 8. nh = hidden_g + pos_attn  (into pk)
    k_add<<<ew_blocks, 256, 0, stream>>>(hidden_g, pq, pk);
    float* nh = pk;

    // 9. hs / h_n ; mix
    k_hsn<<<Bn, Hn, 0, stream>>>(mask, nh, hs, hn);
    k_mix<<<(int)((BH + 255) / 256), 256, 0, stream>>>(hs, hn, W2, W3, W4, mixb);

    // 10. T = nh @ W1^T  (into pv; W1 broadcast via stride 0; K=Hn)
    wmma_gemm<true, 4><<<gemmMHN, 32, 0, stream>>>(
        nh, W1, pv, Hn, Hn, Hn, Hn,
        (long long)Sp * Hn, 0LL, (long long)Sp * Hn, 1.0f);

    // 11. c[b,s] = <lrelu(T + b1 + mix), w_2>
    k_cvec<<<(int)BS, Hn, 0, stream>>>(pv, b1, mixb, w_2, cvec);

    // 12. out = sum_s mask*(c + b_2)*hidden_g
    k_out<<<Bn, Hn, 0, stream>>>(mask, cvec, b_2, hidden_g, out);
}